// DecoderRNN_29283087024480
// MI455X (gfx1250) — compile-verified
//
#include <hip/hip_runtime.h>
#include <hip/hip_bf16.h>
#include <math.h>

#define HID 1024
#define MAXLEN 64
#define VOCAB 50257

typedef __attribute__((ext_vector_type(2))) float v2f;
typedef __attribute__((ext_vector_type(8))) float v8f;

// ---------------------------------------------------------------------------
// WMMA GEMV: y[r] = act( dot(W[r,:], x) + bias[r] ),  r in [0, rows)
// One wave owns a 16-row tile; V_WMMA_F32_16X16X4_F32 accumulates K in steps
// of 4 with TWO interleaved accumulator chains for XDL pipelining.
// B is the x-chunk broadcast across all 16 N-columns (from LDS), so D[:,0]
// holds the 16 dot products.
// ACT: 0 = none, 1 = relu.  NT: non-temporal weight stream (for out_W).
// Requires K % 8 == 0 (here K is 1024 or 2048).
// ---------------------------------------------------------------------------
template <int ACT, bool NT>
__global__ __launch_bounds__(256) void gemv_wmma(const float* __restrict__ W,
                                                 const float* __restrict__ xg,
                                                 const float* __restrict__ bias,
                                                 float* __restrict__ y,
                                                 int rows, int K) {
  __shared__ float xs[2048];  // K <= 2048
  for (int i = threadIdx.x; i < K; i += 256) xs[i] = xg[i];
  __syncthreads();

  const int wave = threadIdx.x >> 5;
  const int lane = threadIdx.x & 31;
  const int tile = blockIdx.x * 8 + wave;
  const int r0 = tile * 16;
  if (r0 >= rows) return;  // wave-uniform

  const int m  = lane & 15;        // M index of this lane's A rows
  const int kk = (lane >> 4) * 2;  // K-pair: lanes 0-15 -> k{0,1}, 16-31 -> k{2,3}
  int rowIdx = r0 + m;
  if (rowIdx >= rows) rowIdx = rows - 1;  // clamp (branch-free; extra rows not stored)
  const float* wrow = W + (size_t)rowIdx * (size_t)K + kk;

  v8f acc0 = {0.f, 0.f, 0.f, 0.f, 0.f, 0.f, 0.f, 0.f};
  v8f acc1 = {0.f, 0.f, 0.f, 0.f, 0.f, 0.f, 0.f, 0.f};
#pragma unroll 4
  for (int k = 0; k < K; k += 8) {
    v2f a0, a1;
    if (NT) {
      a0 = __builtin_nontemporal_load((const v2f*)(wrow + k));
      a1 = __builtin_nontemporal_load((const v2f*)(wrow + k + 4));
    } else {
      a0 = *(const v2f*)(wrow + k);
      a1 = *(const v2f*)(wrow + k + 4);
    }
    v2f b0 = *(const v2f*)(&xs[k + kk]);      // broadcast x-chunk (B cols equal)
    v2f b1 = *(const v2f*)(&xs[k + 4 + kk]);
    acc0 = __builtin_amdgcn_wmma_f32_16x16x4_f32(false, a0, false, b0, (short)0,
                                                 acc0, false, false);
    acc1 = __builtin_amdgcn_wmma_f32_16x16x4_f32(false, a1, false, b1, (short)0,
                                                 acc1, false, false);
  }
  v8f acc = acc0 + acc1;

  // D layout (f32 16x16): VGPR r, lanes 0-15 -> M=r, lanes 16-31 -> M=r+8.
  // Column N=0 lives in lanes 0 and 16.
  if (m == 0) {
    const int base = r0 + (lane >> 4) * 8;
#pragma unroll
    for (int r = 0; r < 8; ++r) {
      const int g = base + r;
      if (g < rows) {
        float v = acc[r] + bias[g];
        if (ACT == 1) v = fmaxf(v, 0.f);
        y[g] = v;
      }
    }
  }
}

// xcat1 = [ emb[token], h ]   (2048 floats)
__global__ void build_xcat1(const int* __restrict__ tok,
                            const float* __restrict__ emb,
                            const float* __restrict__ hidden,
                            float* __restrict__ xcat) {
  int i = blockIdx.x * 256 + threadIdx.x;  // grid 8 x 256 = 2048
  if (i < HID)
    xcat[i] = emb[(size_t)tok[0] * HID + i];
  else if (i < 2 * HID)
    xcat[i] = hidden[i - HID];
}

// softmax over 64 attention logits -> ws copy + d_out copy
__global__ void softmax64(const float* __restrict__ logit,
                          float* __restrict__ aw, float* __restrict__ out_aw) {
  __shared__ float s[64];
  const int t = threadIdx.x;  // 64 threads
  float v = logit[t];
  s[t] = v;
  __syncthreads();
  for (int o = 32; o > 0; o >>= 1) {
    if (t < o) s[t] = fmaxf(s[t], s[t + o]);
    __syncthreads();
  }
  float mx = s[0];
  __syncthreads();
  float e = expf(v - mx);
  s[t] = e;
  __syncthreads();
  for (int o = 32; o > 0; o >>= 1) {
    if (t < o) s[t] += s[t + o];
    __syncthreads();
  }
  float w = e / s[0];
  aw[t] = w;
  out_aw[t] = w;
}

// xcat2 = [ emb[token], attn_weights @ enc_outputs ]
__global__ void attn_apply(const float* __restrict__ aw,
                           const float* __restrict__ enc,
                           const int* __restrict__ tok,
                           const float* __restrict__ emb,
                           float* __restrict__ xcat2) {
  int j = blockIdx.x * 256 + threadIdx.x;  // grid 4 x 256 = 1024
  float s = 0.f;
#pragma unroll 8
  for (int r = 0; r < MAXLEN; ++r) s += aw[r] * enc[r * HID + j];
  xcat2[HID + j] = s;
  xcat2[j] = emb[(size_t)tok[0] * HID + j];
}

// GRU gate math (PyTorch order r,z,n)
__global__ void gru_gates(const float* __restrict__ gi,
                          const float* __restrict__ gh,
                          const float* __restrict__ h,
                          float* __restrict__ hn_ws, float* __restrict__ out_h) {
  int j = blockIdx.x * 256 + threadIdx.x;  // grid 4 x 256 = 1024
  float r = 1.f / (1.f + expf(-(gi[j] + gh[j])));
  float z = 1.f / (1.f + expf(-(gi[j + HID] + gh[j + HID])));
  float n = tanhf(gi[j + 2 * HID] + r * gh[j + 2 * HID]);
  float hn = (1.f - z) * n + z * h[j];
  hn_ws[j] = hn;
  out_h[j] = hn;
}

// single-block log-sum-exp over 50257 logits (L2-hot second pass)
__global__ __launch_bounds__(1024) void lse_kernel(const float* __restrict__ logits,
                                                   float* __restrict__ lse) {
  __shared__ float s[1024];
  const int t = threadIdx.x;
  float m = -INFINITY;
  for (int v = t; v < VOCAB; v += 1024) m = fmaxf(m, logits[v]);
  s[t] = m;
  __syncthreads();
  for (int o = 512; o > 0; o >>= 1) {
    if (t < o) s[t] = fmaxf(s[t], s[t + o]);
    __syncthreads();
  }
  const float gm = s[0];
  __syncthreads();
  float sum = 0.f;
  for (int v = t; v < VOCAB; v += 1024) sum += expf(logits[v] - gm);
  s[t] = sum;
  __syncthreads();
  for (int o = 512; o > 0; o >>= 1) {
    if (t < o) s[t] += s[t + o];
    __syncthreads();
  }
  if (t == 0) *lse = gm + logf(s[0]);
}

__global__ void sub_lse(float* __restrict__ logp, const float* __restrict__ lse) {
  int i = blockIdx.x * 1024 + threadIdx.x;  // grid 50 x 1024
  if (i < VOCAB) logp[i] -= *lse;
}

extern "C" void kernel_launch(void* const* d_in, const int* in_sizes, int n_in,
                              void* d_out, int out_size, void* d_ws, size_t ws_size,
                              hipStream_t stream) {
  const int*   tok     = (const int*)d_in[0];
  const float* hidden  = (const float*)d_in[1];
  const float* enc     = (const float*)d_in[2];
  const float* emb     = (const float*)d_in[3];
  const float* attn_W  = (const float*)d_in[4];
  const float* attn_b  = (const float*)d_in[5];
  const float* comb_W  = (const float*)d_in[6];
  const float* comb_b  = (const float*)d_in[7];
  const float* gru_Wih = (const float*)d_in[8];
  const float* gru_Whh = (const float*)d_in[9];
  const float* gru_bih = (const float*)d_in[10];
  const float* gru_bhh = (const float*)d_in[11];
  const float* out_W   = (const float*)d_in[12];
  const float* out_b   = (const float*)d_in[13];

  float* out = (float*)d_out;            // [0,50257) logp | [50257,51281) h_new | [51281,51345) attn_w
  float* ws  = (float*)d_ws;
  float* xcat1 = ws;                     // 2048
  float* lg64  = ws + 2048;              // 64
  float* aw    = ws + 2112;              // 64
  float* xcat2 = ws + 2176;              // 2048
  float* xx    = ws + 4224;              // 1024
  float* gi    = ws + 5248;              // 3072
  float* gh    = ws + 8320;              // 3072
  float* hn    = ws + 11392;             // 1024
  float* lse   = ws + 12416;             // 1

  build_xcat1<<<8, 256, 0, stream>>>(tok, emb, hidden, xcat1);
  gemv_wmma<0, false><<<1, 256, 0, stream>>>(attn_W, xcat1, attn_b, lg64, MAXLEN, 2 * HID);
  softmax64<<<1, 64, 0, stream>>>(lg64, aw, out + VOCAB + HID);
  attn_apply<<<4, 256, 0, stream>>>(aw, enc, tok, emb, xcat2);
  gemv_wmma<1, false><<<8, 256, 0, stream>>>(comb_W, xcat2, comb_b, xx, HID, 2 * HID);
  gemv_wmma<0, false><<<24, 256, 0, stream>>>(gru_Wih, xx, gru_bih, gi, 3 * HID, HID);
  gemv_wmma<0, false><<<24, 256, 0, stream>>>(gru_Whh, hidden, gru_bhh, gh, 3 * HID, HID);
  gru_gates<<<4, 256, 0, stream>>>(gi, gh, hidden, hn, out + VOCAB);
  gemv_wmma<0, true><<<393, 256, 0, stream>>>(out_W, hn, out_b, out, VOCAB, HID);
  lse_kernel<<<1, 1024, 0, stream>>>(out, lse);
  sub_lse<<<50, 1024, 0, stream>>>(out, lse);
}